// Model_Net_77455440216739
// MI455X (gfx1250) — compile-verified
//
#include <hip/hip_runtime.h>
#include <hip/hip_bf16.h>

#define N_NODES 50000
#define N_EDGES 800000
#define NL      4
#define DIM     128
#define HID     64
#define BATCH   8192
#define NDIS    (BATCH * NL)

typedef float v2f __attribute__((ext_vector_type(2)));
typedef float v8f __attribute__((ext_vector_type(8)));

__device__ __forceinline__ v8f wmma4(v2f a, v2f b, v8f c) {
  // V_WMMA_F32_16X16X4_F32: D(16x16 f32) = A(16x4) * B(4x16) + C
  return __builtin_amdgcn_wmma_f32_16x16x4_f32(false, a, false, b, (short)0, c,
                                               false, false);
}

// ---------------------------------------------------------------- utilities
__global__ void fill_f32(float* __restrict__ p, float v, int n) {
  int i = blockIdx.x * blockDim.x + threadIdx.x;
  if (i < n) p[i] = v;
}

// deg[dst] += 1 per edge (dinv buffer pre-filled with 1.0 for self-loops)
__global__ void deg_kernel(const int* __restrict__ ei, float* __restrict__ deg) {
  int e = blockIdx.x * blockDim.x + threadIdx.x;
  if (e < N_EDGES) atomicAdd(&deg[ei[N_EDGES + e]], 1.0f);
}

__global__ void dinv_kernel(float* __restrict__ d) {
  int n = blockIdx.x * blockDim.x + threadIdx.x;
  if (n < N_NODES) d[n] = rsqrtf(fmaxf(d[n], 1.0f));
}

// ------------------------------------------------ conv1: h = A_norm (x@W1) + b1
// self-loop + bias init: h[n][c] = x[n]*w1[c]*dinv[n]^2 + b1[c]
__global__ void conv1_init(const float* __restrict__ x, const float* __restrict__ w1,
                           const float* __restrict__ b1, const float* __restrict__ dinv,
                           float* __restrict__ h) {
  int gid = blockIdx.x * blockDim.x + threadIdx.x;
  if (gid >= N_NODES * HID) return;
  int n = gid >> 6, c = gid & 63;
  float d = dinv[n];
  h[gid] = x[n] * w1[c] * d * d + b1[c];
}

// per edge: h[dst][c] += x[src]*w1[c]*dinv[src]*dinv[dst]  (16 threads/edge, 4 ch each)
__global__ void conv1_edge(const int* __restrict__ ei, const float* __restrict__ x,
                           const float* __restrict__ w1, const float* __restrict__ dinv,
                           float* __restrict__ h) {
  int t = blockIdx.x * blockDim.x + threadIdx.x;
  int e = t >> 4, q = t & 15;
  if (e >= N_EDGES) return;
  int src = ei[e], dst = ei[N_EDGES + e];
  float coef = x[src] * dinv[src] * dinv[dst];
  float4 w = *(const float4*)(w1 + q * 4);
  float* o = h + (size_t)dst * HID + q * 4;
  atomicAdd(o + 0, coef * w.x);
  atomicAdd(o + 1, coef * w.y);
  atomicAdd(o + 2, coef * w.z);
  atomicAdd(o + 3, coef * w.w);
}

// -------------------------------------------------------------- batch norm
__global__ void bn_stats(const float* __restrict__ h, float* __restrict__ stat) {
  __shared__ float ss[256], sq[256];
  int c = blockIdx.x;  // channel 0..63
  float s = 0.f, q = 0.f;
  for (int n = threadIdx.x; n < N_NODES; n += 256) {
    float v = h[(size_t)n * HID + c];
    s += v; q += v * v;
  }
  ss[threadIdx.x] = s; sq[threadIdx.x] = q;
  __syncthreads();
  for (int off = 128; off; off >>= 1) {
    if (threadIdx.x < off) {
      ss[threadIdx.x] += ss[threadIdx.x + off];
      sq[threadIdx.x] += sq[threadIdx.x + off];
    }
    __syncthreads();
  }
  if (threadIdx.x == 0) {
    float mu = ss[0] * (1.0f / N_NODES);
    float var = sq[0] * (1.0f / N_NODES) - mu * mu;
    stat[c] = mu;
    stat[64 + c] = rsqrtf(var + 1e-5f);
  }
}

__global__ void bn_apply(float* __restrict__ h, const float* __restrict__ stat,
                         const float* __restrict__ gamma, const float* __restrict__ beta) {
  int gid = blockIdx.x * blockDim.x + threadIdx.x;
  if (gid >= N_NODES * HID) return;
  int c = gid & 63;
  float v = (h[gid] - stat[c]) * stat[64 + c] * gamma[c] + beta[c];
  h[gid] = fmaxf(v, 0.f);
}

// --------------------------------------------- h2 = h(50000x64) @ w2(64x128), WMMA f32
__global__ void gemm_h2(const float* __restrict__ h, const float* __restrict__ w2,
                        float* __restrict__ h2) {
  int wave = blockIdx.x * (blockDim.x >> 5) + (threadIdx.x >> 5);
  int lane = threadIdx.x & 31;
  int hf = lane >> 4;        // half-wave: K group / M group for C
  int mrow = lane & 15;      // A row / B,C column
  int tileN = wave & 7;      // 8 N-tiles
  int tileM = wave >> 3;     // 3125 M-tiles
  const float* arow = h + (size_t)(tileM * 16 + mrow) * HID + 2 * hf;
  const float* bcol = w2 + (2 * hf) * DIM + tileN * 16 + mrow;
  v8f acc = {};
  #pragma unroll
  for (int k0 = 0; k0 < HID; k0 += 4) {
    v2f a = *(const v2f*)(arow + k0);
    v2f b; b.x = bcol[k0 * DIM]; b.y = bcol[k0 * DIM + DIM];
    acc = wmma4(a, b, acc);
  }
  float* out = h2 + (size_t)(tileM * 16 + 8 * hf) * DIM + tileN * 16 + mrow;
  #pragma unroll
  for (int v = 0; v < 8; ++v) out[(size_t)v * DIM] = acc[v];
}

// ------------------------------------------------ conv2 init + edge scatter
__global__ void conv2_init(const float* __restrict__ h2, const float* __restrict__ b2,
                           const float* __restrict__ dinv, float* __restrict__ out) {
  int gid = blockIdx.x * blockDim.x + threadIdx.x;
  if (gid >= N_NODES * DIM) return;
  int n = gid >> 7, c = gid & 127;
  float d = dinv[n];
  out[gid] = h2[gid] * d * d + b2[c];
}

__global__ void conv2_edge(const int* __restrict__ ei, const float* __restrict__ h2,
                           const float* __restrict__ dinv, float* __restrict__ out) {
  int t = blockIdx.x * blockDim.x + threadIdx.x;
  int e = t >> 5, q = t & 31;
  if (e >= N_EDGES) return;
  int src = ei[e], dst = ei[N_EDGES + e];
  float coef = dinv[src] * dinv[dst];
  float4 v = *(const float4*)(h2 + (size_t)src * DIM + q * 4);
  float* o = out + (size_t)dst * DIM + q * 4;
  atomicAdd(o + 0, coef * v.x);
  atomicAdd(o + 1, coef * v.y);
  atomicAdd(o + 2, coef * v.z);
  atomicAdd(o + 3, coef * v.w);
}

// ---- G1: dis_mid = relu(gather(32768x256) @ g_w1(256x16) + b), gathered in-kernel
__global__ void gemm_g1(const float* __restrict__ embeds, const int* __restrict__ dis_input,
                        const float* __restrict__ g_w1, const float* __restrict__ g_b1,
                        float* __restrict__ dis_mid) {
  int wave = blockIdx.x * (blockDim.x >> 5) + (threadIdx.x >> 5);  // tileM 0..2047
  int lane = threadIdx.x & 31;
  int hf = lane >> 4, mrow = lane & 15;
  int r = wave * 16 + mrow;
  int ii = dis_input[r * 3 + 0];
  int jj = dis_input[r * 3 + 1];
  int li = dis_input[r * 3 + 2];
  const float* base = embeds + (size_t)li * N_NODES * DIM;
  const float* pi = base + (size_t)ii * DIM;
  const float* pj = base + (size_t)jj * DIM;
  v8f acc = {};
  for (int k0 = 0; k0 < 2 * DIM; k0 += 4) {
    int k = k0 + 2 * hf;  // k and k+1 always fall in the same 128-half
    v2f a = (k < DIM) ? *(const v2f*)(pi + k) : *(const v2f*)(pj + k - DIM);
    v2f b; b.x = g_w1[k * 16 + mrow]; b.y = g_w1[(k + 1) * 16 + mrow];
    acc = wmma4(a, b, acc);
  }
  float bias = g_b1[mrow];
  float* out = dis_mid + (size_t)(wave * 16 + 8 * hf) * 16 + mrow;
  #pragma unroll
  for (int v = 0; v < 8; ++v) out[(size_t)v * 16] = fmaxf(acc[v] + bias, 0.f);
}

// ---- G2: dis_out = dis_mid(32768x16) @ g_w2(16x256) + g_b2
__global__ void gemm_g2(const float* __restrict__ dis_mid, const float* __restrict__ g_w2,
                        const float* __restrict__ g_b2, float* __restrict__ dis_out) {
  int wave = blockIdx.x * (blockDim.x >> 5) + (threadIdx.x >> 5);  // 0..32767
  int lane = threadIdx.x & 31;
  int hf = lane >> 4, mrow = lane & 15;
  int tileN = wave & 15;
  int tileM = wave >> 4;
  const float* arow = dis_mid + (size_t)(tileM * 16 + mrow) * 16 + 2 * hf;
  const float* bcol = g_w2 + (2 * hf) * 256 + tileN * 16 + mrow;
  v8f acc = {};
  #pragma unroll
  for (int k0 = 0; k0 < 16; k0 += 4) {
    v2f a = *(const v2f*)(arow + k0);
    v2f b; b.x = bcol[k0 * 256]; b.y = bcol[k0 * 256 + 256];
    acc = wmma4(a, b, acc);
  }
  float bias = g_b2[tileN * 16 + mrow];
  float* out = dis_out + (size_t)(tileM * 16 + 8 * hf) * 256 + tileN * 16 + mrow;
  #pragma unroll
  for (int v = 0; v < 8; ++v) out[(size_t)v * 256] = acc[v] + bias;
}

// ---- discriminant_out = dis_out @ d_w(256x4) + d_b
__global__ void disc_kernel(const float* __restrict__ dis_out, const float* __restrict__ d_w,
                            const float* __restrict__ d_b, float* __restrict__ out) {
  int gid = blockIdx.x * blockDim.x + threadIdx.x;
  if (gid >= NDIS * 4) return;
  int j = gid & 3, r = gid >> 2;
  const float* row = dis_out + (size_t)r * 256;
  float acc = d_b[j];
  for (int c = 0; c < 256; ++c) acc += row[c] * d_w[c * 4 + j];
  out[gid] = acc;
}

// ---- final head: softmax gate + prediction (one wave32 per row)
__global__ void head_kernel(const float* __restrict__ emb0, const float* __restrict__ node_embed,
                            const int* __restrict__ leftnode, const int* __restrict__ rightnode,
                            const int* __restrict__ indices, const float* __restrict__ dis_out,
                            const float* __restrict__ ws_w, const float* __restrict__ ws_b,
                            const float* __restrict__ p_w, const float* __restrict__ p_b,
                            float* __restrict__ out_pred) {
  int wave = (blockIdx.x * blockDim.x + threadIdx.x) >> 5;
  int lane = threadIdx.x & 31;
  if (wave >= BATCH) return;
  int ln = leftnode[wave], rn = rightnode[wave];
  const float* el = emb0 + (size_t)ln * DIM;
  const float* er = emb0 + (size_t)rn * DIM;
  const float* nl = node_embed + (size_t)ln * DIM;
  const float* nr = node_embed + (size_t)rn * DIM;
  // logits = [node_l, node_r, spec_l, spec_r](512) @ ws_w(512x2)
  float a0 = 0.f, a1 = 0.f;
  for (int c = lane; c < 4 * DIM; c += 32) {
    float v;
    if (c < DIM)            v = nl[c];
    else if (c < 2 * DIM)   v = nr[c - DIM];
    else if (c < 3 * DIM)   v = el[c - 2 * DIM];
    else                    v = er[c - 3 * DIM];
    a0 += v * ws_w[c * 2 + 0];
    a1 += v * ws_w[c * 2 + 1];
  }
  for (int off = 16; off; off >>= 1) { a0 += __shfl_xor(a0, off); a1 += __shfl_xor(a1, off); }
  float l0 = a0 + ws_b[0], l1 = a1 + ws_b[1];
  float mx = fmaxf(l0, l1);
  float e0 = expf((l0 - mx) * 0.5f), e1 = expf((l1 - mx) * 0.5f);  // TEMP = 2
  float inv = 1.0f / (e0 + e1);
  float w0 = e0 * inv, w1 = e1 * inv;
  const float* com = dis_out + (size_t)indices[wave] * 256;
  float p0 = 0.f, p1 = 0.f;
  for (int c = lane; c < 2 * DIM; c += 32) {
    float sp = (c < DIM) ? el[c] : er[c - DIM];
    float pi = sp * w0 + com[c] * w1;
    p0 += pi * p_w[c * 2 + 0];
    p1 += pi * p_w[c * 2 + 1];
  }
  for (int off = 16; off; off >>= 1) { p0 += __shfl_xor(p0, off); p1 += __shfl_xor(p1, off); }
  if (lane == 0) {
    out_pred[wave * 2 + 0] = p0 + p_b[0];
    out_pred[wave * 2 + 1] = p1 + p_b[1];
  }
}

// --------------------------------------------------------------------------
extern "C" void kernel_launch(void* const* d_in, const int* in_sizes, int n_in,
                              void* d_out, int out_size, void* d_ws, size_t ws_size,
                              hipStream_t stream) {
  const int*   leftnode   = (const int*)  d_in[0];
  const int*   rightnode  = (const int*)  d_in[1];
  const int*   dis_input  = (const int*)  d_in[2];
  const int*   indices    = (const int*)  d_in[3];
  const float* gcn_x      = (const float*)d_in[4];
  const int*   edge_index = (const int*)  d_in[5];
  const float* gcn_w1     = (const float*)d_in[6];
  const float* gcn_b1     = (const float*)d_in[7];
  const float* bn_gamma   = (const float*)d_in[8];
  const float* bn_beta    = (const float*)d_in[9];
  const float* gcn_w2     = (const float*)d_in[10];
  const float* gcn_b2     = (const float*)d_in[11];
  const float* g_w1       = (const float*)d_in[12];
  const float* g_b1       = (const float*)d_in[13];
  const float* g_w2       = (const float*)d_in[14];
  const float* g_b2       = (const float*)d_in[15];
  const float* ws_w       = (const float*)d_in[16];
  const float* ws_b       = (const float*)d_in[17];
  const float* d_w        = (const float*)d_in[18];
  const float* d_b        = (const float*)d_in[19];
  const float* p_w        = (const float*)d_in[20];
  const float* p_b        = (const float*)d_in[21];
  const float* node_embed = (const float*)d_in[22];

  float* ws      = (float*)d_ws;
  float* embeds  = ws;                                      // 4*50000*128
  float* h       = embeds + (size_t)NL * N_NODES * DIM;     // 50000*64
  float* h2      = h + (size_t)N_NODES * HID;               // 50000*128
  float* dinv    = h2 + (size_t)N_NODES * DIM;              // 50000
  float* stat    = dinv + N_NODES;                          // 128
  float* dis_mid = stat + 128;                              // 32768*16
  float* dis_out = dis_mid + (size_t)NDIS * 16;             // 32768*256

  for (int l = 0; l < NL; ++l) {
    const int*   ei  = edge_index + (size_t)l * 2 * N_EDGES;
    const float* xl  = gcn_x    + (size_t)l * N_NODES;
    const float* w1  = gcn_w1   + (size_t)l * HID;
    const float* b1  = gcn_b1   + (size_t)l * HID;
    const float* gam = bn_gamma + (size_t)l * HID;
    const float* bet = bn_beta  + (size_t)l * HID;
    const float* w2  = gcn_w2   + (size_t)l * HID * DIM;
    const float* b2  = gcn_b2   + (size_t)l * DIM;
    float*       emb = embeds   + (size_t)l * N_NODES * DIM;

    fill_f32  <<<(N_NODES + 255) / 256, 256, 0, stream>>>(dinv, 1.0f, N_NODES);
    deg_kernel<<<(N_EDGES + 255) / 256, 256, 0, stream>>>(ei, dinv);
    dinv_kernel<<<(N_NODES + 255) / 256, 256, 0, stream>>>(dinv);
    conv1_init<<<(N_NODES * HID) / 256, 256, 0, stream>>>(xl, w1, b1, dinv, h);
    conv1_edge<<<(N_EDGES * 16) / 256, 256, 0, stream>>>(ei, xl, w1, dinv, h);
    bn_stats  <<<HID, 256, 0, stream>>>(h, stat);
    bn_apply  <<<(N_NODES * HID) / 256, 256, 0, stream>>>(h, stat, gam, bet);
    gemm_h2   <<<(N_NODES / 16) * (DIM / 16) / 8, 256, 0, stream>>>(h, w2, h2);
    conv2_init<<<(N_NODES * DIM) / 256, 256, 0, stream>>>(h2, b2, dinv, emb);
    conv2_edge<<<(N_EDGES * 32) / 256, 256, 0, stream>>>(ei, h2, dinv, emb);
  }

  gemm_g1<<<(NDIS / 16) / 8, 256, 0, stream>>>(embeds, dis_input, g_w1, g_b1, dis_mid);
  gemm_g2<<<(NDIS / 16) * 16 / 8, 256, 0, stream>>>(dis_mid, g_w2, g_b2, dis_out);
  disc_kernel<<<(NDIS * 4) / 256, 256, 0, stream>>>(dis_out, d_w, d_b,
                                                    (float*)d_out + BATCH * 2);
  head_kernel<<<(BATCH * 32) / 256, 256, 0, stream>>>(embeds, node_embed, leftnode,
                                                      rightnode, indices, dis_out,
                                                      ws_w, ws_b, p_w, p_b,
                                                      (float*)d_out);
}